// weightedSDTW_57758720196766
// MI455X (gfx1250) — compile-verified
//
#include <hip/hip_runtime.h>
#include <math.h>

typedef __attribute__((ext_vector_type(2))) float v2f;
typedef __attribute__((ext_vector_type(8))) float v8f;

#define B_ 64
#define M_ 512
#define N_ 512
#define D_ 64

// ---------------------------------------------------------------------------
// Kernel 1: squared norms x2[b][n] = ||X[b,n,:]||^2, y2[b][m] = ||Y[b,m,:]||^2
// ---------------------------------------------------------------------------
__global__ void sdtw_norms_kernel(const float* __restrict__ X,
                                  const float* __restrict__ Y,
                                  float* __restrict__ x2,
                                  float* __restrict__ y2) {
    int v = blockIdx.x * blockDim.x + threadIdx.x;   // 0 .. B*(N+M)-1
    const int nx = B_ * N_;
    const float* src;
    float* dst;
    int idx;
    if (v < nx) { src = X; dst = x2; idx = v; }
    else        { src = Y; dst = y2; idx = v - nx; }
    const float4* p = (const float4*)(src + (size_t)idx * D_);
    float s = 0.f;
#pragma unroll
    for (int k = 0; k < D_ / 4; ++k) {
        float4 q = p[k];
        s += q.x * q.x + q.y * q.y + q.z * q.z + q.w * q.w;
    }
    dst[idx] = s;
}

// ---------------------------------------------------------------------------
// Kernel 2: cost matrix via fp32 WMMA.
// One wave (32 threads) per 16x16 tile: C[b,m,n] = y2[m] + x2[n] - 2*<Y_m,X_n>
// K=64 accumulated as 16 chained V_WMMA_F32_16X16X4_F32.
// ---------------------------------------------------------------------------
__global__ void sdtw_cost_wmma_kernel(const float* __restrict__ X,
                                      const float* __restrict__ Y,
                                      const float* __restrict__ x2,
                                      const float* __restrict__ y2,
                                      float* __restrict__ C) {
    const int lane = threadIdx.x;        // 0..31, block == one wave32
    const int n0 = blockIdx.x * 16;
    const int m0 = blockIdx.y * 16;
    const int b  = blockIdx.z;

    const int l15 = lane & 15;
    const int hi  = lane >> 4;           // 0 for lanes 0-15, 1 for lanes 16-31

    // A (16x4, f32): lane -> row M = l15; lanes 0-15 hold K = k0,k0+1 ;
    //                lanes 16-31 hold K = k0+2,k0+3  (per ISA 32-bit A layout)
    const float* Yrow = Y + (((size_t)b * M_) + (size_t)(m0 + l15)) * D_;
    // B (4x16, f32): row-striped across lanes within each VGPR:
    //   VGPR0: lanes 0-15 K=k0, lanes 16-31 K=k0+1 ; VGPR1: K=k0+2 / k0+3
    const float* Xrow = X + (((size_t)b * N_) + (size_t)(n0 + l15)) * D_;

    v8f acc = {};
#pragma unroll
    for (int k0 = 0; k0 < D_; k0 += 4) {
        v2f a = *(const v2f*)(Yrow + k0 + hi * 2);   // aligned 8B load
        v2f bb;
        bb.x = Xrow[k0 + hi];
        bb.y = Xrow[k0 + 2 + hi];
        acc = __builtin_amdgcn_wmma_f32_16x16x4_f32(
            /*neg_a=*/false, a, /*neg_b=*/false, bb,
            /*c_mod=*/(short)0, acc, /*reuse_a=*/false, /*reuse_b=*/false);
    }

    // C/D layout: VGPR r, lanes 0-15 -> M=r, N=lane ; lanes 16-31 -> M=r+8
    const int n = n0 + l15;
    const float x2v = x2[(size_t)b * N_ + n];
#pragma unroll
    for (int r = 0; r < 8; ++r) {
        const int m = m0 + r + 8 * hi;
        const float c = y2[(size_t)b * M_ + m] + x2v - 2.0f * acc[r];
        C[(((size_t)b * M_) + m) * N_ + n] = c;
    }
}

// ---------------------------------------------------------------------------
// Kernel 3: weighted soft-DTW wavefront DP. One block per batch, one thread
// per diagonal cell, 3 rolling diagonals in LDS, one barrier per diagonal.
// ---------------------------------------------------------------------------
__global__ void __launch_bounds__(512)
sdtw_dp_kernel(const float* __restrict__ C, float* __restrict__ bres) {
    const int b   = blockIdx.x;
    const int tid = threadIdx.x;     // 0..511
    const int i   = tid + 1;         // DP row index 1..M
    __shared__ float diag[3][M_ + 8];   // diag[p%3][i] = D[i][p - i]

    const float INF = __builtin_inff();
    // sum==0 diagonal: D[0][0]=0, rest inf.  sum==1 diagonal: all inf.
    diag[0][i] = INF;
    diag[1][i] = INF;
    if (tid == 0) { diag[0][0] = 0.0f; diag[1][0] = INF; }
    __syncthreads();

    const float* Cb = C + (size_t)b * M_ * N_;
    const float w0 = 0.5f, w1 = 1.5f, w2 = 1.0f;

    for (int t = 0; t < M_ + N_ - 1; ++t) {
        float*       cur = diag[(t + 2) % 3];   // diagonal sum t+2 (overwrites t-1)
        const float* p1  = diag[(t + 1) % 3];   // sum t+1
        const float* p2  = diag[t % 3];         // sum t
        const int j = t + 2 - i;
        float dnew = INF;
        if (j >= 1 && j <= N_) {
            const float c  = Cb[(size_t)(i - 1) * N_ + (j - 1)];
            const float a0 = fmaf(c, w2, p2[i - 1]);  // diagonal move
            const float a1 = fmaf(c, w0, p1[i - 1]);  // from (i-1, j)
            const float a2 = fmaf(c, w1, p1[i]);      // from (i, j-1)
            const float mu = fminf(a0, fminf(a1, a2));
            const float s  = expf(mu - a0) + expf(mu - a1) + expf(mu - a2);
            dnew = mu - logf(s);                      // -gamma * LSE(-a/gamma), gamma=1
        }
        cur[i] = dnew;
        if (tid == 0) cur[0] = INF;                   // D[0][t+2] border
        __syncthreads();
    }
    // D[M][N] lives on diagonal sum M+N at row index M
    if (tid == 0) bres[b] = diag[(M_ + N_) % 3][M_];
}

// ---------------------------------------------------------------------------
// Kernel 4: mean over batches -> d_out[0]
// ---------------------------------------------------------------------------
__global__ void sdtw_reduce_kernel(const float* __restrict__ bres,
                                   float* __restrict__ out) {
    if (threadIdx.x == 0) {
        float s = 0.f;
        for (int b = 0; b < B_; ++b) s += bres[b];
        out[0] = s * (1.0f / B_);
    }
}

// ---------------------------------------------------------------------------
extern "C" void kernel_launch(void* const* d_in, const int* in_sizes, int n_in,
                              void* d_out, int out_size, void* d_ws, size_t ws_size,
                              hipStream_t stream) {
    const float* X = (const float*)d_in[0];   // (B, N, d)
    const float* Y = (const float*)d_in[1];   // (B, M, d)
    float* out = (float*)d_out;

    char* ws = (char*)d_ws;
    float* x2   = (float*)(ws);                                   // B*N floats
    float* y2   = (float*)(ws + (size_t)B_ * N_ * sizeof(float)); // B*M floats
    float* bres = (float*)(ws + (size_t)(B_ * N_ + B_ * M_) * sizeof(float));
    float* C    = (float*)(ws + (size_t)512 * 1024);              // 64 MB (L2-resident)

    sdtw_norms_kernel<<<dim3((B_ * (M_ + N_)) / 256), 256, 0, stream>>>(X, Y, x2, y2);
    sdtw_cost_wmma_kernel<<<dim3(N_ / 16, M_ / 16, B_), 32, 0, stream>>>(X, Y, x2, y2, C);
    sdtw_dp_kernel<<<dim3(B_), 512, 0, stream>>>(C, bres);
    sdtw_reduce_kernel<<<dim3(1), 64, 0, stream>>>(bres, out);
}